// SRNLayer_3788161155574
// MI455X (gfx1250) — compile-verified
//
#include <hip/hip_runtime.h>
#include <hip/hip_bf16.h>
#include <math.h>

typedef float v2f __attribute__((ext_vector_type(2)));
typedef float v8f __attribute__((ext_vector_type(8)));

#define KSZ   5
#define PADC  2
#define EPSN  1e-4f
#define DEG   4
#define B_    4
#define C_    3
#define H_    96
#define O_    32
#define S_    2
#define N_    75      // C*KSZ*KSZ (feature dim)
#define P_    9       // 1 + S*DEG
#define L_    9216    // H*H
#define KPAD  80      // padded K for A rows (valid 0..74, zeros beyond)
#define NROW  80      // padded feature rows for staged Y
#define MP    16      // P padded to 16 -> one o == one M tile
#define LDSTR 84      // LDS row stride for A tile (84 mod 64 = 20 -> conflict-free)

// ---- workspace layout (float offsets) ----
#define WS_A      0                          // [32][16][80]      = 40960
#define WS_CINV   (WS_A + O_*MP*KPAD)        // [32][96]          =  3072
#define WS_SUMSQ  (WS_CINV + O_*96)          // [4][9216]         = 36864
#define WS_SCALE  (WS_SUMSQ + B_*L_)         // [4][9216]         = 36864
#define WS_Y      (WS_SCALE + B_*L_)         // [4][80][9216]     = 2949120
// total = 3,066,880 floats ~= 12.3 MB

// ------------------------------------------------------------------
// Kernel 1: build design matrix Xf, padded A = Xf^T, cov^-1 (9x9 GJ)
// ------------------------------------------------------------------
__global__ void srn_prep_kernel(const float* __restrict__ weights,
                                float* __restrict__ ws) {
    int o = threadIdx.x;
    if (o >= O_) return;

    float Xf[N_][P_];
    for (int n = 0; n < N_; ++n) {
        float x0 = weights[o * N_ * S_ + n * S_ + 0];
        float x1 = weights[o * N_ * S_ + n * S_ + 1];
        Xf[n][0] = 1.0f;
        float p0 = 1.0f, p1 = 1.0f;
        for (int d = 0; d < DEG; ++d) {
            p0 *= x0; p1 *= x1;
            Xf[n][1 + d * 2 + 0] = p0;
            Xf[n][1 + d * 2 + 1] = p1;
        }
    }

    // A rows: r = o*16 + p, cols k = feature n (zero padded rows/cols)
    float* A = ws + WS_A;
    for (int p = 0; p < MP; ++p)
        for (int k = 0; k < KPAD; ++k)
            A[(o * MP + p) * KPAD + k] = (p < P_ && k < N_) ? Xf[k][p] : 0.0f;

    // cov = Xf^T Xf  (9x9)
    float cov[P_][P_];
    for (int p = 0; p < P_; ++p)
        for (int q = 0; q < P_; ++q) cov[p][q] = 0.0f;
    for (int n = 0; n < N_; ++n)
        for (int p = 0; p < P_; ++p)
            for (int q = 0; q < P_; ++q)
                cov[p][q] = fmaf(Xf[n][p], Xf[n][q], cov[p][q]);

    // Gauss-Jordan inverse with partial pivoting
    float M[P_][2 * P_];
    for (int r = 0; r < P_; ++r)
        for (int c = 0; c < P_; ++c) {
            M[r][c]      = cov[r][c];
            M[r][P_ + c] = (r == c) ? 1.0f : 0.0f;
        }
    for (int col = 0; col < P_; ++col) {
        int piv = col;
        float best = fabsf(M[col][col]);
        for (int r = col + 1; r < P_; ++r) {
            float a = fabsf(M[r][col]);
            if (a > best) { best = a; piv = r; }
        }
        if (piv != col)
            for (int c = 0; c < 2 * P_; ++c) {
                float t = M[col][c]; M[col][c] = M[piv][c]; M[piv][c] = t;
            }
        float inv = 1.0f / M[col][col];
        for (int c = 0; c < 2 * P_; ++c) M[col][c] *= inv;
        for (int r = 0; r < P_; ++r) {
            if (r == col) continue;
            float f = M[r][col];
            for (int c = 0; c < 2 * P_; ++c)
                M[r][c] = fmaf(-f, M[col][c], M[r][c]);
        }
    }
    float* Ci = ws + WS_CINV + o * 96;
    for (int p = 0; p < P_; ++p)
        for (int q = 0; q < P_; ++q)
            Ci[p * P_ + q] = M[p][P_ + q];
}

// ------------------------------------------------------------------
// Kernel 2: fused unfold + eps*noise -> Y[b][n][l], plus per-(b,l)
// sumsq and 1/(75*var) for the ddof=1 std normalization.
// ------------------------------------------------------------------
__global__ void srn_unfold_kernel(const float* __restrict__ y2,
                                  const float* __restrict__ noise,
                                  float* __restrict__ ws) {
    int idx = blockIdx.x * blockDim.x + threadIdx.x;
    if (idx >= B_ * L_) return;
    int b = idx / L_;
    int l = idx - b * L_;
    int h = l / H_;
    int w = l - h * H_;

    float* Y = ws + WS_Y + (size_t)b * NROW * L_;
    float sum = 0.0f, sumsq = 0.0f;
    int n = 0;
    for (int c = 0; c < C_; ++c) {
        for (int i = 0; i < KSZ; ++i) {
            int hh = h + i - PADC;
            bool hin = (hh >= 0) && (hh < H_);
            for (int j = 0; j < KSZ; ++j, ++n) {
                int ww = w + j - PADC;
                float v = 0.0f;
                if (hin && ww >= 0 && ww < H_)
                    v = y2[((b * C_ + c) * H_ + hh) * H_ + ww];
                v = fmaf(EPSN, noise[((size_t)b * N_ + n) * L_ + l], v);
                Y[(size_t)n * L_ + l] = v;
                sum += v;
                sumsq = fmaf(v, v, sumsq);
            }
        }
    }
    // zero K padding rows (ws is poisoned by the harness)
    for (int nz = N_; nz < NROW; ++nz) Y[(size_t)nz * L_ + l] = 0.0f;

    float var = (sumsq - sum * sum * (1.0f / (float)N_)) * (1.0f / (float)(N_ - 1));
    ws[WS_SUMSQ + idx] = sumsq;
    ws[WS_SCALE + idx] = 1.0f / ((float)N_ * var);
}

// ------------------------------------------------------------------
// Kernel 3: f32 WMMA GEMM  T = A * Y  (one wave per 16x16 tile,
// one o per M-tile) + projection-residual epilogue.
// ------------------------------------------------------------------
__global__ __launch_bounds__(256) void srn_gemm_kernel(const float* __restrict__ ws,
                                                       float* __restrict__ out) {
    __shared__ float sA[MP * LDSTR];
    __shared__ float sCi[P_ * P_];

    const int o = blockIdx.y;
    const int b = blockIdx.z;
    const int tid = threadIdx.x;

    // stage A tile (16x80) and covinv (81) into LDS, bank-conflict-free stride
    const float* Ag = ws + WS_A + (size_t)o * MP * KPAD;
    for (int i = tid; i < MP * KPAD; i += 256) {
        int r = i / KPAD;
        int k = i - r * KPAD;
        sA[r * LDSTR + k] = Ag[i];
    }
    if (tid < P_ * P_) sCi[tid] = ws[WS_CINV + o * 96 + tid];
    __syncthreads();

    const int wave = tid >> 5;
    const int lane = tid & 31;
    const int half = lane & 15;
    const int koff = (lane < 16) ? 0 : 2;   // upper half-wave carries K+2,K+3
    const int l0 = (blockIdx.x * 8 + wave) * 16;

    const float* Yg = ws + WS_Y + (size_t)b * NROW * L_;

    v8f acc = {};
    for (int kk = 0; kk < 19; ++kk) {       // K = 76 (75 valid + 1 zero pad)
        const int k = kk * 4 + koff;
        v2f a, bf;
        a.x  = sA[half * LDSTR + k];
        a.y  = sA[half * LDSTR + k + 1];
        bf.x = Yg[(size_t)k * L_ + l0 + half];
        bf.y = Yg[(size_t)(k + 1) * L_ + l0 + half];
        acc = __builtin_amdgcn_wmma_f32_16x16x4_f32(
            /*neg_a=*/false, a, /*neg_b=*/false, bf,
            /*c_mod=*/(short)0, acc, /*reuse_a=*/false, /*reuse_b=*/false);
    }

    // t_8 for column (l0+lane) lives in lane+16's acc[0]
    float t8 = __shfl_xor(acc[0], 16, 32);

    if (lane < 16) {
        const int l = l0 + lane;
        float t[P_];
        #pragma unroll
        for (int p = 0; p < 8; ++p) t[p] = acc[p];
        t[8] = t8;

        float qf = 0.0f;
        #pragma unroll
        for (int p = 0; p < P_; ++p) {
            float tp = t[p];
            #pragma unroll
            for (int q = 0; q < P_; ++q)
                qf = fmaf(sCi[p * P_ + q] * tp, t[q], qf);
        }

        const float sumsq = ws[WS_SUMSQ + b * L_ + l];
        const float scale = ws[WS_SCALE + b * L_ + l];
        const float err = (sumsq - qf) * scale;   // = mean_n residual^2 of normalized y

        const float e1 = 0.36787944117144233f;    // exp(-1)
        const float inv1me1 = 1.0f / (1.0f - e1);
        out[((size_t)(b * O_) + o) * L_ + l] = (__expf(-err) - e1) * inv1me1 - 0.5f;
    }
}

// ------------------------------------------------------------------
extern "C" void kernel_launch(void* const* d_in, const int* in_sizes, int n_in,
                              void* d_out, int out_size, void* d_ws, size_t ws_size,
                              hipStream_t stream) {
    const float* y2      = (const float*)d_in[0];   // [4,3,96,96]
    const float* weights = (const float*)d_in[1];   // [32,3,5,5,2]
    const float* noise   = (const float*)d_in[2];   // [4,75,9216]
    float* out = (float*)d_out;                     // [4,32,96,96]
    float* ws  = (float*)d_ws;

    hipLaunchKernelGGL(srn_prep_kernel, dim3(1), dim3(32), 0, stream, weights, ws);

    const int total = B_ * L_;
    hipLaunchKernelGGL(srn_unfold_kernel, dim3((total + 255) / 256), dim3(256), 0,
                       stream, y2, noise, ws);

    // grid: x = 576/8 N-tiles, y = o (32), z = b (4); 8 waves/block, 1 tile/wave
    hipLaunchKernelGGL(srn_gemm_kernel, dim3((L_ / 16) / 8, O_, B_), dim3(256), 0,
                       stream, ws, out);
}